// BackWarp_tf_2130303779364
// MI455X (gfx1250) — compile-verified
//
#include <hip/hip_runtime.h>
#include <hip/hip_bf16.h>
#include <stdint.h>

// Backward warp (bilinear gather), B=16, H=W=1024, C=3, float32.
// ~512 MB moved, ~0.67 GFLOP -> bandwidth/VALU bound; no WMMA applies.
// MI455X tuning: 4 pixels/thread; NT b128 flow loads + NT b128 output stores
// (keep the 192MB image resident in the 192MB L2); corner gathers via GVS
// addressing (uniform SGPR image base + 32-bit lane offsets, image < 2GB) with
// corner-delta offsets; branchless CDNA5 global_prefetch_b8 on the flow.

#define HH 1024
#define WW 1024
#define CC 3
#define HW (HH * WW)   // 1 << 20

typedef float v4f __attribute__((ext_vector_type(4)));

__device__ __forceinline__ void warp_one(const char* __restrict__ image,  // uniform base
                                         uint32_t base12,                  // b*H*W*12 bytes
                                         float xg, float yg,
                                         float fx, float fy,
                                         float& o0, float& o1, float& o2)
{
    float x = xg + fx;
    float y = yg + fy;

    // Faithful normalization round-trip from the reference (identity up to
    // float rounding; same evaluation order).
    x = 0.5f * (2.0f * (x / (float)WW - 0.5f) + 1.0f) * (float)WW;
    y = 0.5f * (2.0f * (y / (float)HH - 0.5f) + 1.0f) * (float)HH;

    // Truncation toward zero == astype(int32).
    int x0 = (int)x;
    int x1 = x0 + 1;
    int y0 = (int)y;
    int y1 = y0 + 1;
    x0 = min(max(x0, 0), WW - 1);
    x1 = min(max(x1, 0), WW - 1);
    y0 = min(max(y0, 0), HH - 1);
    y1 = min(max(y1, 0), HH - 1);

    // Bilinear weights from the *clipped* corner coordinates (as reference).
    float wxa = (float)x1 - x;
    float wxb = x - (float)x0;
    float wya = (float)y1 - y;
    float wyb = y - (float)y0;
    float wa = wxa * wya;   // (y0,x0)
    float wb = wxa * wyb;   // (y1,x0)
    float wc = wxb * wya;   // (y0,x1)
    float wd = wxb * wyb;   // (y1,x1)

    // 32-bit GVS byte offsets: one full chain + deltas (x1-x0, y1-y0 in {0,1}).
    uint32_t offA = base12 + (uint32_t)((y0 << 10) | x0) * 12u;   // (y0,x0)
    uint32_t dy   = (uint32_t)(y1 - y0) * (uint32_t)(WW * CC * 4); // 0 or 12288
    uint32_t dx   = (uint32_t)(x1 - x0) * 12u;                     // 0 or 12
    uint32_t offB = offA + dy;        // (y1,x0)
    uint32_t offC = offA + dx;        // (y0,x1)
    uint32_t offD = offB + dx;        // (y1,x1)

    const float* pa = (const float*)(image + offA);
    const float* pb = (const float*)(image + offB);
    const float* pc = (const float*)(image + offC);
    const float* pd = (const float*)(image + offD);

    float a0 = pa[0], a1 = pa[1], a2 = pa[2];
    float b0 = pb[0], b1 = pb[1], b2 = pb[2];
    float c0 = pc[0], c1 = pc[1], c2 = pc[2];
    float d0 = pd[0], d1 = pd[1], d2 = pd[2];

    o0 = wa * a0 + wb * b0 + wc * c0 + wd * d0;
    o1 = wa * a1 + wb * b1 + wc * c1 + wd * d1;
    o2 = wa * a2 + wb * b2 + wc * c2 + wd * d2;
}

__global__ __launch_bounds__(256) void backwarp_kernel(
    const float* __restrict__ image,
    const float* __restrict__ flow,
    float* __restrict__ out,
    int nquad)  // (B*H*W)/4 : each thread handles 4 consecutive pixels
{
    int t = blockIdx.x * blockDim.x + threadIdx.x;
    if (t >= nquad) return;

    int i0 = t << 2;   // pixel index multiple of 4; quad never straddles a row

    // Branchless CDNA5 prefetch (global_prefetch_b8) of the flow stream
    // ~64KB ahead; clamp instead of branching (redundant prefetch is free).
    int pf = min(i0 + 8192, 4 * nquad - 4);
    __builtin_prefetch(flow + (size_t)pf * 2, /*rw=*/0, /*locality=*/0);

    // Two 16-byte non-temporal loads per lane: flow for all 4 pixels
    // (b128, TH=NT -> the 128MB flow stream does not evict the image).
    v4f f01 = __builtin_nontemporal_load((const v4f*)flow + 2 * (size_t)t);
    v4f f23 = __builtin_nontemporal_load((const v4f*)flow + 2 * (size_t)t + 1);

    // Shared grid decomposition for the quad (H = W = 1024, 4 | W).
    int xi    = i0 & (WW - 1);          // multiple of 4 -> xi+3 in same row
    int yi    = (i0 >> 10) & (HH - 1);
    int bbase = i0 & ~(HW - 1);         // b * H * W
    uint32_t base12 = (uint32_t)bbase * 12u;   // byte offset of this batch image
    const char* imgc = (const char*)image;
    float yg = (float)yi;

    float o[12];
    warp_one(imgc, base12, (float)xi,       yg, f01.x, f01.y, o[0], o[1],  o[2]);
    warp_one(imgc, base12, (float)(xi + 1), yg, f01.z, f01.w, o[3], o[4],  o[5]);
    warp_one(imgc, base12, (float)(xi + 2), yg, f23.x, f23.y, o[6], o[7],  o[8]);
    warp_one(imgc, base12, (float)(xi + 3), yg, f23.z, f23.w, o[9], o[10], o[11]);

    // 48 contiguous write-once bytes at u32 byte offset t*48 (16B aligned):
    // exactly three NT b128 stores in GVS form.
    v4f s0 = {o[0], o[1], o[2],  o[3]};
    v4f s1 = {o[4], o[5], o[6],  o[7]};
    v4f s2 = {o[8], o[9], o[10], o[11]};
    v4f* po = (v4f*)((char*)out + (uint32_t)t * 48u);
    __builtin_nontemporal_store(s0, po + 0);
    __builtin_nontemporal_store(s1, po + 1);
    __builtin_nontemporal_store(s2, po + 2);
}

extern "C" void kernel_launch(void* const* d_in, const int* in_sizes, int n_in,
                              void* d_out, int out_size, void* d_ws, size_t ws_size,
                              hipStream_t stream) {
    const float* image = (const float*)d_in[0];  // [B,H,W,3] float32
    const float* flow  = (const float*)d_in[1];  // [B,H,W,2] float32
    float*       out   = (float*)d_out;          // [B,H,W,3] float32

    int npix  = in_sizes[1] / 2;       // B*H*W (flow: 2 floats per pixel)
    int nquad = npix / 4;              // B*H*W = 16M, divisible by 4
    int threads = 256;                 // 8 wave32s per block
    int blocks  = (nquad + threads - 1) / threads;

    backwarp_kernel<<<blocks, threads, 0, stream>>>(image, flow, out, nquad);
}